// ArapProject_46059229282958
// MI455X (gfx1250) — compile-verified
//
#include <hip/hip_runtime.h>
#include <math.h>

// ============================================================================
// ARAP projection, 8 Adam steps. N=100k vertices, K=8 CSR neighbors.
//
// Layout strategy (working set is L2-resident; kernels are gather-latency
// bound, so minimize load transactions with aligned 128-bit gathers):
//   V buffer : per-vertex 32B record {xyz.x,y,z,pad, recon.x,y,z,pad}
//              -> neighbor geometry = 2x global_load_b128
//   R buffer : per-vertex 48B row (9 floats + 3 pad)
//              -> neighbor rotation = 3x global_load_b128
//   edges    : K==8 fast path loads indices/weights as 2x int4 + 2x float4
// recon half of V ping-pongs V0/V1; step 8 writes d_out [N,3] directly.
// Workspace: V0[8N] | V1[8N] | R[12N] | m[3N] | v[3N] = 34N floats (13.6 MB).
// ============================================================================

#ifndef ARAP_POLAR_ITERS
#define ARAP_POLAR_ITERS 8
#endif

__global__ __launch_bounds__(256) void arap_init_kernel(
    const float* __restrict__ xyz, const float* __restrict__ recon0,
    float4* __restrict__ V0, float4* __restrict__ V1,
    float* __restrict__ m, float* __restrict__ v, int n)
{
    int i = blockIdx.x * blockDim.x + threadIdx.x;
    if (i >= n) return;
    const float x  = xyz[3*i+0],    y  = xyz[3*i+1],    z  = xyz[3*i+2];
    const float rx = recon0[3*i+0], ry = recon0[3*i+1], rz = recon0[3*i+2];
    const float4 p = make_float4(x, y, z, 0.0f);
    V0[2*i+0] = p;
    V0[2*i+1] = make_float4(rx, ry, rz, 0.0f);
    V1[2*i+0] = p;
    V1[2*i+1] = make_float4(0.0f, 0.0f, 0.0f, 0.0f);  // overwritten by step 1
    m[3*i+0] = 0.0f; m[3*i+1] = 0.0f; m[3*i+2] = 0.0f;
    v[3*i+0] = 0.0f; v[3*i+1] = 0.0f; v[3*i+2] = 0.0f;
}

// --------------------------------------------------------------------------
// Rotation kernel: S_i = sum_j w_ij (x_i-x_j)(x'_i-x'_j)^T
// R_i = polar(S_i^T) via Higham scaled Newton (== V U^T of SVD(S), det>0)
// --------------------------------------------------------------------------
__global__ __launch_bounds__(256) void arap_rot_kernel(
    const float4* __restrict__ V,
    const int*    __restrict__ nbrs, const int* __restrict__ numN,
    const int*    __restrict__ accN, const float* __restrict__ wmat,
    float4* __restrict__ Rout, int n)
{
    int i = blockIdx.x * blockDim.x + threadIdx.x;
    if (i >= n) return;

    const float4 pa = V[2*i+0];
    const float4 pb = V[2*i+1];
    const float xix = pa.x, xiy = pa.y, xiz = pa.z;
    const float dix = pb.x, diy = pb.y, diz = pb.z;
    const int acc = accN[i];
    const int num = numN[i];

    float s00=0.f,s01=0.f,s02=0.f,s10=0.f,s11=0.f,s12=0.f,s20=0.f,s21=0.f,s22=0.f;

    auto edge = [&](int nb, float w) {
        const float4 na  = V[2*nb+0];
        const float4 nbv = V[2*nb+1];
        const float ex = xix - na.x,  ey = xiy - na.y,  ez = xiz - na.z;
        const float dx = dix - nbv.x, dy = diy - nbv.y, dz = diz - nbv.z;
        const float wx = w*ex, wy = w*ey, wz = w*ez;
        s00 = fmaf(wx, dx, s00); s01 = fmaf(wx, dy, s01); s02 = fmaf(wx, dz, s02);
        s10 = fmaf(wy, dx, s10); s11 = fmaf(wy, dy, s11); s12 = fmaf(wy, dz, s12);
        s20 = fmaf(wz, dx, s20); s21 = fmaf(wz, dy, s21); s22 = fmaf(wz, dz, s22);
    };

    if (num == 8 && (acc & 3) == 0) {               // K==8 fast path (aligned)
        const int4*   ip = (const int4*)(nbrs + acc);
        const float4* wp = (const float4*)(wmat + acc);
        const int4   ia = ip[0], ib = ip[1];
        const float4 wa = wp[0], wb = wp[1];
        edge(ia.x, wa.x); edge(ia.y, wa.y); edge(ia.z, wa.z); edge(ia.w, wa.w);
        edge(ib.x, wb.x); edge(ib.y, wb.y); edge(ib.z, wb.z); edge(ib.w, wb.w);
    } else {
        for (int j = 0; j < num; ++j) edge(nbrs[acc + j], wmat[acc + j]);
    }

    // X = S^T ; R = polar(X): X <- 0.5(g X + (1/g) X^-T)
    float x00=s00, x01=s10, x02=s20;
    float x10=s01, x11=s11, x12=s21;
    float x20=s02, x21=s12, x22=s22;

#pragma unroll
    for (int it = 0; it < ARAP_POLAR_ITERS; ++it) {
        const float a0 = x11*x22 - x21*x12;   // c1 x c2
        const float a1 = x21*x02 - x01*x22;
        const float a2 = x01*x12 - x11*x02;
        const float b0 = x12*x20 - x22*x10;   // c2 x c0
        const float b1 = x22*x00 - x02*x20;
        const float b2 = x02*x10 - x12*x00;
        const float c0 = x10*x21 - x20*x11;   // c0 x c1
        const float c1 = x20*x01 - x00*x21;
        const float c2 = x00*x11 - x10*x01;

        float det = fmaf(x00, a0, fmaf(x10, a1, x20*a2));
        const float ad = fabsf(det);
        det = (ad < 1e-30f) ? ((det < 0.0f) ? -1e-30f : 1e-30f) : det;
        const float invd = 1.0f / det;

        const float y00 = a0*invd, y10 = a1*invd, y20 = a2*invd;
        const float y01 = b0*invd, y11 = b1*invd, y21 = b2*invd;
        const float y02 = c0*invd, y12 = c1*invd, y22 = c2*invd;

        const float fx = x00*x00+x01*x01+x02*x02+x10*x10+x11*x11+x12*x12
                       + x20*x20+x21*x21+x22*x22;
        const float fy = y00*y00+y01*y01+y02*y02+y10*y10+y11*y11+y12*y12
                       + y20*y20+y21*y21+y22*y22;
        float g = sqrtf(sqrtf(fy / fx));          // (||X^-1||/||X||)^(1/2)
        g = fminf(fmaxf(g, 1e-4f), 1e4f);
        const float ig = 1.0f / g;

        x00 = 0.5f*fmaf(g, x00, ig*y00); x01 = 0.5f*fmaf(g, x01, ig*y01); x02 = 0.5f*fmaf(g, x02, ig*y02);
        x10 = 0.5f*fmaf(g, x10, ig*y10); x11 = 0.5f*fmaf(g, x11, ig*y11); x12 = 0.5f*fmaf(g, x12, ig*y12);
        x20 = 0.5f*fmaf(g, x20, ig*y20); x21 = 0.5f*fmaf(g, x21, ig*y21); x22 = 0.5f*fmaf(g, x22, ig*y22);
    }

    Rout[3*i+0] = make_float4(x00, x01, x02, x10);
    Rout[3*i+1] = make_float4(x11, x12, x20, x21);
    Rout[3*i+2] = make_float4(x22, 0.0f, 0.0f, 0.0f);
}

// --------------------------------------------------------------------------
// Gradient + Adam kernel:
//   g_i = aw * sum_j w_ij (2(p'_i-p'_j) - (R_i+R_j)(x_i-x_j))
// packOut != nullptr on the final step -> write [N,3] result to d_out.
// --------------------------------------------------------------------------
__global__ __launch_bounds__(256) void arap_grad_adam_kernel(
    const float4* __restrict__ Vin,
    const int*    __restrict__ nbrs, const int* __restrict__ numN,
    const int*    __restrict__ accN, const float* __restrict__ wmat,
    const float4* __restrict__ Rm,   const float* __restrict__ arapW,
    float* __restrict__ m, float* __restrict__ v,
    float4* __restrict__ Vout, float* __restrict__ packOut,
    int n, float corr1, float corr2)
{
    int i = blockIdx.x * blockDim.x + threadIdx.x;
    if (i >= n) return;

    const float4 pa = Vin[2*i+0];
    const float4 pb = Vin[2*i+1];
    const float xix = pa.x, xiy = pa.y, xiz = pa.z;
    const float dix = pb.x, diy = pb.y, diz = pb.z;

    const float4 ra = Rm[3*i+0];
    const float4 rb = Rm[3*i+1];
    const float4 rc = Rm[3*i+2];
    const float r00=ra.x, r01=ra.y, r02=ra.z;
    const float r10=ra.w, r11=rb.x, r12=rb.y;
    const float r20=rb.z, r21=rb.w, r22=rc.x;

    const int acc = accN[i];
    const int num = numN[i];

    float gx = 0.f, gy = 0.f, gz = 0.f;

    auto edge = [&](int nb, float w) {
        const float4 na  = Vin[2*nb+0];
        const float4 nbv = Vin[2*nb+1];
        const float ex = xix - na.x,  ey = xiy - na.y,  ez = xiz - na.z;
        const float dx = dix - nbv.x, dy = diy - nbv.y, dz = diz - nbv.z;

        const float4 sa = Rm[3*nb+0];
        const float4 sb = Rm[3*nb+1];
        const float4 sc = Rm[3*nb+2];
        const float q00 = r00 + sa.x, q01 = r01 + sa.y, q02 = r02 + sa.z;
        const float q10 = r10 + sa.w, q11 = r11 + sb.x, q12 = r12 + sb.y;
        const float q20 = r20 + sb.z, q21 = r21 + sb.w, q22 = r22 + sc.x;

        const float tx = 2.0f*dx - fmaf(q00, ex, fmaf(q01, ey, q02*ez));
        const float ty = 2.0f*dy - fmaf(q10, ex, fmaf(q11, ey, q12*ez));
        const float tz = 2.0f*dz - fmaf(q20, ex, fmaf(q21, ey, q22*ez));

        gx = fmaf(w, tx, gx);
        gy = fmaf(w, ty, gy);
        gz = fmaf(w, tz, gz);
    };

    if (num == 8 && (acc & 3) == 0) {               // K==8 fast path (aligned)
        const int4*   ip = (const int4*)(nbrs + acc);
        const float4* wp = (const float4*)(wmat + acc);
        const int4   ia = ip[0], ib = ip[1];
        const float4 wa = wp[0], wb = wp[1];
        edge(ia.x, wa.x); edge(ia.y, wa.y); edge(ia.z, wa.z); edge(ia.w, wa.w);
        edge(ib.x, wb.x); edge(ib.y, wb.y); edge(ib.z, wb.z); edge(ib.w, wb.w);
    } else {
        for (int j = 0; j < num; ++j) edge(nbrs[acc + j], wmat[acc + j]);
    }

    const float aw = arapW[0];
    gx *= aw; gy *= aw; gz *= aw;

    const float B1 = 0.9f, B2 = 0.999f;
    const float OB1 = 1.0f - 0.9f, OB2 = 1.0f - 0.999f;

    float mx = m[3*i+0], my = m[3*i+1], mz = m[3*i+2];
    float vx = v[3*i+0], vy = v[3*i+1], vz = v[3*i+2];

    mx = fmaf(B1, mx, OB1*gx); my = fmaf(B1, my, OB1*gy); mz = fmaf(B1, mz, OB1*gz);
    vx = fmaf(B2, vx, OB2*gx*gx); vy = fmaf(B2, vy, OB2*gy*gy); vz = fmaf(B2, vz, OB2*gz*gz);

    m[3*i+0] = mx; m[3*i+1] = my; m[3*i+2] = mz;
    v[3*i+0] = vx; v[3*i+1] = vy; v[3*i+2] = vz;

    const float mhx = mx / corr1, mhy = my / corr1, mhz = mz / corr1;
    const float vhx = vx / corr2, vhy = vy / corr2, vhz = vz / corr2;

    const float RATE = 0.01f, EPS = 1e-9f;
    const float nx = dix - RATE * (mhx / (sqrtf(vhx) + EPS));
    const float ny = diy - RATE * (mhy / (sqrtf(vhy) + EPS));
    const float nz = diz - RATE * (mhz / (sqrtf(vhz) + EPS));

    if (packOut != nullptr) {                     // final step: [N,3] output
        packOut[3*i+0] = nx;
        packOut[3*i+1] = ny;
        packOut[3*i+2] = nz;
    } else {
        Vout[2*i+1] = make_float4(nx, ny, nz, 0.0f);   // single b128 store
    }
}

extern "C" void kernel_launch(void* const* d_in, const int* in_sizes, int n_in,
                              void* d_out, int out_size, void* d_ws, size_t ws_size,
                              hipStream_t stream) {
    (void)n_in; (void)out_size; (void)ws_size;

    const float* xyz    = (const float*)d_in[0];
    const float* recon0 = (const float*)d_in[1];
    const int*   nbrs   = (const int*)  d_in[2];
    const int*   numN   = (const int*)  d_in[3];
    const int*   accN   = (const int*)  d_in[4];
    const float* wmat   = (const float*)d_in[5];
    const float* arapW  = (const float*)d_in[6];

    const int n = in_sizes[0] / 3;   // vertices

    float*  ws   = (float*)d_ws;
    float4* V0   = (float4*)ws;                       // 2N float4 (8N floats)
    float4* V1   = V0 + (size_t)2 * n;                // 2N float4
    float4* Rbuf = (float4*)(ws + (size_t)16 * n);    // 3N float4 (12N floats)
    float*  mbuf = ws + (size_t)28 * n;               // 3N
    float*  vbuf = ws + (size_t)31 * n;               // 3N   (total 34N floats)

    const int T  = 256;
    const int GV = (n + T - 1) / T;

    arap_init_kernel<<<GV, T, 0, stream>>>(xyz, recon0, V0, V1, mbuf, vbuf, n);

    double b1p = 1.0, b2p = 1.0;
    const int NUMSTEPS = 8;
    for (int s = 1; s <= NUMSTEPS; ++s) {
        b1p *= 0.9; b2p *= 0.999;
        const float c1 = (float)(1.0 - b1p);
        const float c2 = (float)(1.0 - b2p);

        const float4* Vin  = (s & 1) ? V0 : V1;  // step1 reads V0
        float4*       Vout = (s & 1) ? V1 : V0;
        float* packOut = (s == NUMSTEPS) ? (float*)d_out : nullptr;

        arap_rot_kernel<<<GV, T, 0, stream>>>(Vin, nbrs, numN, accN,
                                              wmat, Rbuf, n);
        arap_grad_adam_kernel<<<GV, T, 0, stream>>>(Vin, nbrs, numN, accN,
                                                    wmat, Rbuf, arapW,
                                                    mbuf, vbuf, Vout, packOut,
                                                    n, c1, c2);
    }
}